// Model_22024592294336
// MI455X (gfx1250) — compile-verified
//
#include <hip/hip_runtime.h>
#include <hip/hip_bf16.h>
#include <math.h>

// ---------------- problem constants ----------------
#define B0    8
#define C0    32
#define T0    512
#define PRED  192
#define PL    16
#define MM    8
#define EL    2
#define SS    32
#define RR    8
#define MODES 32
#define DIN   128          // M*PL
#define PP    32           // T/PL
#define PADL  7            // (M-1)*DELAY
#define BCN   256          // B0*C0
#define NPROJ 72           // R + 2S
#define NPROJ_PAD 80

// ---------------- LDS padded strides (64 banks, conflict-free frag loads) ---
#define PSTR  132          // 132 % 64 = 4  -> 16 lanes x4 banks distinct
#define DBSTR 84           // 84 % 64 = 20  -> distinct
#define HSTR  33           // odd -> distinct
#define XKSTR 561          // 17*33, 561%64=49 -> distinct
#define ZSTR  68           // 68 % 64 = 4   -> distinct

typedef float v2f __attribute__((ext_vector_type(2)));
typedef float v8f __attribute__((ext_vector_type(8)));

// D(16x16 f32) = A(16x4 f32) * B(4x16 f32) + C   -- fp32 matrix pipe
__device__ __forceinline__ v8f wmma4(v2f a, v2f b, v8f c) {
  return __builtin_amdgcn_wmma_f32_16x16x4_f32(false, a, false, b, (short)0, c,
                                               false, false);
}

// ~306KB dynamic LDS per workgroup (<= 320KB WGP budget; 1 WG/WGP -> huge VGPR
// headroom per wave, which we spend on multi-accumulator WMMA chains)
struct ResLDS {
  float enc[PP * PSTR];         // residual input
  float xn [PP * PSTR];         // layernorm output
  float dlt[PP * PSTR];         // softplus(delta)
  float dbc[PP * DBSTR];        // [delta_r | Bm | Cm] (80 cols used)
  float An [DIN * HSTR];        // -exp(A_log)
  float h  [DIN * HSTR];        // scan state
  float Xr [MODES * XKSTR];     // rfft real, [k][p_half(17)][i(33)]
  float Xi [MODES * XKSTR];     // rfft imag
  float Z  [PP * ZSTR];         // [Zr | Zi] per p (64 cols used)
  float Fc [DIN * MODES];       // forward DFT cos
  float Fs [DIN * MODES];       // forward DFT -sin
  float G  [2 * MODES * DIN];   // irfft basis [cos rows | -sin rows]
};

// ============ kernel 1: instance norm + PSR embedding ============
__global__ __launch_bounds__(128) void prep_kernel(const float* __restrict__ x,
                                                   float* __restrict__ enc_g,
                                                   float* __restrict__ stats) {
  __shared__ float red[128];
  const int bc = blockIdx.x;
  const int b = bc >> 5, c = bc & 31;
  const float* xb = x + (size_t)b * T0 * C0 + c;
  const int tid = threadIdx.x;
  float s = 0.f, s2 = 0.f;
  for (int t = tid; t < T0; t += 128) {
    float v = xb[(size_t)t * C0];
    s += v; s2 += v * v;
  }
  red[tid] = s; __syncthreads();
  for (int off = 64; off >= 1; off >>= 1) { if (tid < off) red[tid] += red[tid + off]; __syncthreads(); }
  const float mean = red[0] * (1.f / T0);
  __syncthreads();
  red[tid] = s2; __syncthreads();
  for (int off = 64; off >= 1; off >>= 1) { if (tid < off) red[tid] += red[tid + off]; __syncthreads(); }
  const float var = red[0] * (1.f / T0) - mean * mean;
  const float sd = sqrtf(var + 1e-5f);
  const float rstd = 1.f / sd;
  if (tid == 0) { stats[bc] = mean; stats[BCN + bc] = sd; }
  float* eo = enc_g + (size_t)bc * (PP * DIN);
  // enc[bc, p, m*16+j] = xn[b, p*16+j-7+m, c]  (zero in the left pad)
  for (int idx = tid; idx < PP * DIN; idx += 128) {
    const int p = idx >> 7, d = idx & 127;
    const int m = d >> 4, j = d & 15;
    const int t = p * PL + j;
    float v = 0.f;
    if (t >= PADL) v = (xb[(size_t)(t - PADL + m) * C0] - mean) * rstd;
    eo[idx] = v;
  }
}

// ============ kernel 2: weight pre-transposes (coalesced WMMA B-operands) ====
__global__ void init_kernel(const float* __restrict__ xw, const float* __restrict__ dw,
                            const float* __restrict__ kr, const float* __restrict__ ki,
                            const float* __restrict__ ow,
                            float* __restrict__ xwT, float* __restrict__ dwT,
                            float* __restrict__ kTr, float* __restrict__ kTi,
                            float* __restrict__ owT) {
  const int gsz = gridDim.x * blockDim.x;
  const int g = blockIdx.x * blockDim.x + threadIdx.x;
  for (int idx = g; idx < EL * DIN * NPROJ_PAD; idx += gsz) {        // [li][d][r]
    const int li = idx / (DIN * NPROJ_PAD), rem = idx % (DIN * NPROJ_PAD);
    const int d = rem / NPROJ_PAD, r = rem % NPROJ_PAD;
    xwT[idx] = (r < NPROJ) ? xw[((size_t)li * NPROJ + r) * DIN + d] : 0.f;
  }
  for (int idx = g; idx < EL * RR * DIN; idx += gsz) {               // [li][r][d]
    const int li = idx / (RR * DIN), rem = idx % (RR * DIN);
    const int r = rem / DIN, d = rem % DIN;
    dwT[idx] = dw[((size_t)li * DIN + d) * RR + r];
  }
  for (int idx = g; idx < EL * MODES * SS * SS; idx += gsz) {        // [li][k][i][o]
    const int li = idx / (MODES * SS * SS), rem = idx % (MODES * SS * SS);
    const int k = rem / (SS * SS), rem2 = rem % (SS * SS);
    const int i = rem2 / SS, o = rem2 % SS;
    const size_t src = (((size_t)li * SS + i) * SS + o) * MODES + k;
    kTr[idx] = kr[src];
    kTi[idx] = ki[src];
  }
  for (int idx = g; idx < (PP * DIN) * PRED; idx += gsz) {           // [j][q]
    const int j = idx / PRED, q = idx % PRED;
    owT[idx] = ow[(size_t)q * (PP * DIN) + j];
  }
}

// ============ kernel 3: fused resblock (one workgroup per bc) ============
__global__ __launch_bounds__(256) void resblock_kernel(
    float* __restrict__ enc_g,
    const float* __restrict__ xwT,   // [DIN][80]
    const float* __restrict__ dwT,   // [8][DIN]
    const float* __restrict__ dtb,   // [DIN]
    const float* __restrict__ alog,  // [DIN][32]
    const float* __restrict__ dpw,   // [DIN]
    const float* __restrict__ kTr,   // [32 modes][32 i][32 o]
    const float* __restrict__ kTi,
    const float* __restrict__ lng,
    const float* __restrict__ lnb) {
  extern __shared__ char smem_raw[];
  ResLDS& L = *reinterpret_cast<ResLDS*>(smem_raw);
  const int tid  = threadIdx.x;
  const int lane = tid & 31;
  const int wave = tid >> 5;
  const int am   = lane & 15;            // A-row / B-col / D-col lane index
  const int akb  = (lane >> 4) << 1;     // k sub-offset (0 or 2)
  const int mhi  = (lane >> 4) << 3;     // D-row base (0 or 8)
  const int bc   = blockIdx.x;
  float* eg = enc_g + (size_t)bc * (PP * DIN);

  // stage 0: load enc, build DFT tables + Aneg, clear scan state
  for (int idx = tid; idx < PP * DIN; idx += 256)
    L.enc[(idx >> 7) * PSTR + (idx & 127)] = eg[idx];
  for (int idx = tid; idx < DIN * MODES; idx += 256) {
    const int d = idx >> 5, k = idx & 31;
    const int dk = (d * k) & 127;                         // exact period-128 reduction
    float sn, cs;
    __sincosf((float)dk * 0.049087385212340526f, &sn, &cs); // 2*pi/128
    L.Fc[d * MODES + k] = cs;
    L.Fs[d * MODES + k] = -sn;
    const float ck = (k == 0) ? (1.f / 128.f) : (2.f / 128.f);
    L.G[k * DIN + d]           = ck * cs;                 // Re path
    L.G[(MODES + k) * DIN + d] = -ck * sn;                // Im path (0 at k=0)
  }
  for (int idx = tid; idx < DIN * SS; idx += 256) {
    const int d = idx >> 5, s = idx & 31;
    L.An[d * HSTR + s] = -__expf(alog[idx]);
    L.h [d * HSTR + s] = 0.f;
  }
  __syncthreads();

  // stage 1: LayerNorm over DIN (each wave owns 4 rows)
  for (int r = 0; r < 4; ++r) {
    const int p = wave * 4 + r;
    float sum = 0.f;
    for (int j = 0; j < 4; ++j) sum += L.enc[p * PSTR + lane * 4 + j];
    for (int m = 16; m >= 1; m >>= 1) sum += __shfl_xor(sum, m, 32);
    const float mu = sum * (1.f / DIN);
    float vs = 0.f;
    for (int j = 0; j < 4; ++j) { float t = L.enc[p * PSTR + lane * 4 + j] - mu; vs += t * t; }
    for (int m = 16; m >= 1; m >>= 1) vs += __shfl_xor(vs, m, 32);
    const float rstd = rsqrtf(vs * (1.f / DIN) + 1e-5f);
    for (int j = 0; j < 4; ++j) {
      const int d = lane * 4 + j;
      L.xn[p * PSTR + d] = (L.enc[p * PSTR + d] - mu) * rstd * lng[d] + lnb[d];
    }
  }
  __syncthreads();

  // stage 2: xproj GEMM  dbc(32x80) = xn(32x128) @ xwT(128x80)
  // 4 independent accumulator chains hide XDL latency (latency-bound kernel).
  for (int t = wave; t < 10; t += 8) {
    const int pb = (t / 5) * 16, rb = (t % 5) * 16;
    auto step = [&](int ks, v8f c) {
      const int k0 = ks * 4 + akb;
      v2f a, b;
      a.x = L.xn[(pb + am) * PSTR + k0];
      a.y = L.xn[(pb + am) * PSTR + k0 + 1];
      b.x = xwT[k0 * NPROJ_PAD + rb + am];
      b.y = xwT[(k0 + 1) * NPROJ_PAD + rb + am];
      return wmma4(a, b, c);
    };
    v8f c0 = {}, c1 = {}, c2 = {}, c3 = {};
    for (int ks = 0; ks < 32; ks += 4) {
      c0 = step(ks + 0, c0);
      c1 = step(ks + 1, c1);
      c2 = step(ks + 2, c2);
      c3 = step(ks + 3, c3);
    }
    const v8f acc = (c0 + c1) + (c2 + c3);
    for (int v = 0; v < 8; ++v)
      L.dbc[(pb + mhi + v) * DBSTR + rb + am] = acc[v];
  }
  __syncthreads();

  // stage 3: dtproj GEMM (K=8, two independent chains) + softplus -> delta
  for (int t = wave; t < 16; t += 8) {
    const int pb = (t >> 3) * 16, db = (t & 7) * 16;
    auto step = [&](int ks, v8f c) {
      const int k0 = ks * 4 + akb;
      v2f a, b;
      a.x = L.dbc[(pb + am) * DBSTR + k0];
      a.y = L.dbc[(pb + am) * DBSTR + k0 + 1];
      b.x = dwT[k0 * DIN + db + am];
      b.y = dwT[(k0 + 1) * DIN + db + am];
      return wmma4(a, b, c);
    };
    v8f c0 = {}, c1 = {};
    c0 = step(0, c0);
    c1 = step(1, c1);
    const v8f acc = c0 + c1;
    const float bias = dtb[db + am];
    for (int v = 0; v < 8; ++v) {
      const float xv = acc[v] + bias;
      L.dlt[(pb + mhi + v) * PSTR + db + am] = (xv > 20.f) ? xv : log1pf(__expf(xv));
    }
  }
  __syncthreads();

  // stage 4: sequential SSM scan over p; per-p forward DFT (GEMM);
  //          per-half (16 p's) batched mode mixing (GEMM) + Cm contraction
  for (int half = 0; half < 2; ++half) {
    for (int pl2 = 0; pl2 < 16; ++pl2) {
      const int p = half * 16 + pl2;
      // h = exp(delta*A)*h + (delta*xn)*Bm   (4096 elems, all 256 threads)
      for (int idx = tid; idx < DIN * SS; idx += 256) {
        const int d = idx >> 5, s = idx & 31;
        const float de = L.dlt[p * PSTR + d];
        const float dA = __expf(de * L.An[d * HSTR + s]);
        const float bx = de * L.xn[p * PSTR + d] * L.dbc[p * DBSTR + RR + s];
        L.h[d * HSTR + s] = dA * L.h[d * HSTR + s] + bx;
      }
      __syncthreads();
      // rfft modes: X(32i x 32k) = h^T(32x128) @ F(128x32); 8 tile jobs = 8 waves
      {
        const int ri = wave >> 2, it = (wave >> 1) & 1, kt = wave & 1;
        const int ib = it * 16, kb = kt * 16;
        const float* Fm = ri ? L.Fs : L.Fc;
        auto step = [&](int ks, v8f c) {
          const int d0 = ks * 4 + akb;
          v2f a, b;
          a.x = L.h[d0 * HSTR + ib + am];
          a.y = L.h[(d0 + 1) * HSTR + ib + am];
          b.x = Fm[d0 * MODES + kb + am];
          b.y = Fm[(d0 + 1) * MODES + kb + am];
          return wmma4(a, b, c);
        };
        v8f c0 = {}, c1 = {}, c2 = {}, c3 = {};
        for (int ks = 0; ks < 32; ks += 4) {
          c0 = step(ks + 0, c0);
          c1 = step(ks + 1, c1);
          c2 = step(ks + 2, c2);
          c3 = step(ks + 3, c3);
        }
        const v8f acc = (c0 + c1) + (c2 + c3);
        float* X = ri ? L.Xi : L.Xr;
        const int k = kb + am;
        for (int v = 0; v < 8; ++v)
          X[k * XKSTR + pl2 * HSTR + ib + mhi + v] = acc[v];
      }
      __syncthreads();
    }
    // mode mixing: per k, Y(16p x 32o) = X(16p x 32i) @ W(32x32) (complex),
    // then Z[p,k] = sum_o Y[p,o]*Cm[p,o] fused from the accumulators.
    // 4 accumulator chains (2 N-tiles x re/im) interleaved; A fragments reused
    // across both N-tiles; next mode's Kw tiles prefetched into L2/L0.
    for (int kq = 0; kq < 4; ++kq) {
      const int k = wave * 4 + kq;
      const float* Wr = kTr + (size_t)k * SS * SS;
      const float* Wi = kTi + (size_t)k * SS * SS;
      if (kq < 3) {   // prefetch next mode's 4KB Wr/Wi (32 lanes x 128B lines)
        __builtin_prefetch(Wr + SS * SS + lane * 32, 0, 3);
        __builtin_prefetch(Wi + SS * SS + lane * 32, 0, 3);
      }
      v8f ar0 = {}, ai0 = {}, ar1 = {}, ai1 = {};
      for (int ks = 0; ks < 8; ++ks) {
        const int i0 = ks * 4 + akb;
        v2f aR, aI;
        aR.x = L.Xr[k * XKSTR + am * HSTR + i0];
        aR.y = L.Xr[k * XKSTR + am * HSTR + i0 + 1];
        aI.x = L.Xi[k * XKSTR + am * HSTR + i0];
        aI.y = L.Xi[k * XKSTR + am * HSTR + i0 + 1];
        v2f bR0, bI0, bR1, bI1, bIn0, bIn1;
        bR0.x = Wr[i0 * SS + am];            bR0.y = Wr[(i0 + 1) * SS + am];
        bI0.x = Wi[i0 * SS + am];            bI0.y = Wi[(i0 + 1) * SS + am];
        bR1.x = Wr[i0 * SS + 16 + am];       bR1.y = Wr[(i0 + 1) * SS + 16 + am];
        bI1.x = Wi[i0 * SS + 16 + am];       bI1.y = Wi[(i0 + 1) * SS + 16 + am];
        bIn0.x = -bI0.x; bIn0.y = -bI0.y;
        bIn1.x = -bI1.x; bIn1.y = -bI1.y;
        ar0 = wmma4(aR, bR0, ar0);   // Yr += Xr*Wr
        ai0 = wmma4(aR, bI0, ai0);   // Yi += Xr*Wi
        ar1 = wmma4(aR, bR1, ar1);
        ai1 = wmma4(aR, bI1, ai1);
        ar0 = wmma4(aI, bIn0, ar0);  // Yr -= Xi*Wi
        ai0 = wmma4(aI, bR0, ai0);   // Yi += Xi*Wr
        ar1 = wmma4(aI, bIn1, ar1);
        ai1 = wmma4(aI, bR1, ai1);
      }
      float zr[8], zi[8];
      for (int v = 0; v < 8; ++v) {
        const int p = half * 16 + mhi + v;
        const float cm0 = L.dbc[p * DBSTR + RR + SS + am];        // Cm[p][o], o<16
        const float cm1 = L.dbc[p * DBSTR + RR + SS + 16 + am];   // o>=16
        zr[v] = ar0[v] * cm0 + ar1[v] * cm1;
        zi[v] = ai0[v] * cm0 + ai1[v] * cm1;
      }
      for (int m = 8; m >= 1; m >>= 1)                // half-wave row reduce
        for (int v = 0; v < 8; ++v) {
          zr[v] += __shfl_xor(zr[v], m, 32);
          zi[v] += __shfl_xor(zi[v], m, 32);
        }
      if (am == 0) {
        for (int v = 0; v < 8; ++v) {
          const int p = half * 16 + mhi + v;
          L.Z[p * ZSTR + k]         = zr[v];
          L.Z[p * ZSTR + MODES + k] = zi[v];
        }
      }
    }
    __syncthreads();
  }

  // stage 5: irfft GEMM  ys(32x128) = Z(32x64) @ G(64x128), + residuals, store
  for (int t = wave; t < 16; t += 8) {
    const int pb = (t >> 3) * 16, db = (t & 7) * 16;
    auto step = [&](int ks, v8f c) {
      const int k0 = ks * 4 + akb;
      v2f a, b;
      a.x = L.Z[(pb + am) * ZSTR + k0];
      a.y = L.Z[(pb + am) * ZSTR + k0 + 1];
      b.x = L.G[k0 * DIN + db + am];
      b.y = L.G[(k0 + 1) * DIN + db + am];
      return wmma4(a, b, c);
    };
    v8f c0 = {}, c1 = {}, c2 = {}, c3 = {};
    for (int ks = 0; ks < 16; ks += 4) {
      c0 = step(ks + 0, c0);
      c1 = step(ks + 1, c1);
      c2 = step(ks + 2, c2);
      c3 = step(ks + 3, c3);
    }
    const v8f acc = (c0 + c1) + (c2 + c3);
    const int d = db + am;
    const float dpv = dpw[d];
    for (int v = 0; v < 8; ++v) {
      const int p = pb + mhi + v;
      eg[p * DIN + d] = acc[v] + L.enc[p * PSTR + d] + dpv * L.xn[p * PSTR + d];
    }
  }
}

// ============ kernel 4: output GEMM 256x4096x192 + denorm epilogue ============
__global__ __launch_bounds__(128) void out_kernel(const float* __restrict__ enc_g,
                                                  const float* __restrict__ owT, // [4096][192]
                                                  const float* __restrict__ ob,
                                                  const float* __restrict__ stats,
                                                  float* __restrict__ out) {
  __shared__ float sA[16 * PSTR];
  const int tid = threadIdx.x, lane = tid & 31, wave = tid >> 5;
  const int am = lane & 15, akb = (lane >> 4) << 1, mhi = (lane >> 4) << 3;
  const int mb = (blockIdx.x / 3) * 16;
  const int qb = ((blockIdx.x % 3) * 4 + wave) * 16;
  v8f c0 = {}, c1 = {}, c2 = {}, c3 = {};   // 4 chains persist across K chunks
  for (int kc = 0; kc < 32; ++kc) {
    __syncthreads();
    for (int idx = tid; idx < 16 * DIN; idx += 128) {
      const int r = idx >> 7, cc = idx & 127;
      sA[r * PSTR + cc] = enc_g[(size_t)(mb + r) * (PP * DIN) + kc * DIN + cc];
    }
    __syncthreads();
    auto step = [&](int ks, v8f c) {
      const int k0 = ks * 4 + akb;
      v2f a, b;
      a.x = sA[am * PSTR + k0];
      a.y = sA[am * PSTR + k0 + 1];
      const size_t kg = (size_t)kc * DIN + k0;
      b.x = owT[kg * PRED + qb + am];
      b.y = owT[(kg + 1) * PRED + qb + am];
      return wmma4(a, b, c);
    };
    for (int ks = 0; ks < 32; ks += 4) {
      c0 = step(ks + 0, c0);
      c1 = step(ks + 1, c1);
      c2 = step(ks + 2, c2);
      c3 = step(ks + 3, c3);
    }
  }
  const v8f acc = (c0 + c1) + (c2 + c3);
  const int q = qb + am;
  const float bias = ob[q];
  for (int v = 0; v < 8; ++v) {
    const int bcq = mb + mhi + v;
    const int bb = bcq >> 5, cc = bcq & 31;
    out[((size_t)bb * PRED + q) * C0 + cc] =
        (acc[v] + bias) * stats[BCN + bcq] + stats[bcq];
  }
}

// ============ host entry ============
extern "C" void kernel_launch(void* const* d_in, const int* in_sizes, int n_in,
                              void* d_out, int out_size, void* d_ws, size_t ws_size,
                              hipStream_t stream) {
  (void)in_sizes; (void)n_in; (void)out_size; (void)ws_size;
  const float* x        = (const float*)d_in[0];
  const float* xproj_w  = (const float*)d_in[4];
  const float* dtproj_w = (const float*)d_in[5];
  const float* dtproj_b = (const float*)d_in[6];
  const float* A_log    = (const float*)d_in[7];
  const float* Dp       = (const float*)d_in[8];
  const float* Kw_r     = (const float*)d_in[9];
  const float* Kw_i     = (const float*)d_in[10];
  const float* ln_g     = (const float*)d_in[11];
  const float* ln_b     = (const float*)d_in[12];
  const float* out_w    = (const float*)d_in[13];
  const float* out_b    = (const float*)d_in[14];

  // workspace layout (floats), total ~7.6 MB
  float* ws      = (float*)d_ws;
  float* d_enc   = ws;  ws += (size_t)BCN * PP * DIN;        // 1,048,576
  float* d_stats = ws;  ws += 2 * BCN;                       // 512
  float* d_xwT   = ws;  ws += (size_t)EL * DIN * NPROJ_PAD;  // 20,480
  float* d_dwT   = ws;  ws += (size_t)EL * RR * DIN;         // 2,048
  float* d_kTr   = ws;  ws += (size_t)EL * MODES * SS * SS;  // 65,536
  float* d_kTi   = ws;  ws += (size_t)EL * MODES * SS * SS;  // 65,536
  float* d_owT   = ws;                                       // 786,432

  hipLaunchKernelGGL(prep_kernel, dim3(BCN), dim3(128), 0, stream, x, d_enc, d_stats);
  hipLaunchKernelGGL(init_kernel, dim3(512), dim3(256), 0, stream,
                     xproj_w, dtproj_w, Kw_r, Kw_i, out_w,
                     d_xwT, d_dwT, d_kTr, d_kTi, d_owT);
  const size_t lds = sizeof(ResLDS);
  for (int li = 0; li < EL; ++li) {
    hipLaunchKernelGGL(resblock_kernel, dim3(BCN), dim3(256), lds, stream,
                       d_enc,
                       d_xwT + (size_t)li * DIN * NPROJ_PAD,
                       d_dwT + (size_t)li * RR * DIN,
                       dtproj_b + (size_t)li * DIN,
                       A_log + (size_t)li * DIN * SS,
                       Dp + (size_t)li * DIN,
                       d_kTr + (size_t)li * MODES * SS * SS,
                       d_kTi + (size_t)li * MODES * SS * SS,
                       ln_g + (size_t)li * DIN,
                       ln_b + (size_t)li * DIN);
  }
  hipLaunchKernelGGL(out_kernel, dim3(48), dim3(128), 0, stream,
                     d_enc, d_owT, out_b, d_stats, (float*)d_out);
}